// HMMEstimator_62912680951977
// MI455X (gfx1250) — compile-verified
//
#include <hip/hip_runtime.h>

typedef float v2f __attribute__((ext_vector_type(2)));
typedef float v8f __attribute__((ext_vector_type(8)));

#define IMG   512
#define TILE  16
#define HALO  18           // TILE + 2 (3x3 SAME conv halo)
#define PITCH 20           // padded row pitch (bank-conflict padding)
#define CHSZ  (HALO*PITCH) // 360 floats per channel plane
#define NCH   30           // 3 (Z) + 27 (w_pre tangents)

// Fused HMM-estimator update:
//  Phase 1: per halo pixel, compute Z and w_pre (30 channels) into LDS.
//  Phase 2: 3x3 conv of 10 groups of 3 channels via V_WMMA_F32_16X16X4_F32,
//           softmax (group 0) + softmax-JVP combine (groups 1..9).
__global__ __launch_bounds__(256) void hmm_fused_kernel(
    const float* __restrict__ obs,     // (512,512,3)
    const float* __restrict__ P,       // (3,3)  b_s = sum_o P[o*3+s]*obs_o
    const float* __restrict__ u_k,     // (512,512,3,1)
    const float* __restrict__ w_k,     // (512,512,3,1,3,3)
    const float* __restrict__ conv_w,  // (3,3,3,3) OIHW
    const float* __restrict__ conv_b,  // (3,)
    float* __restrict__ u_out,         // (512,512,3,1)
    float* __restrict__ w_out)         // (512,512,3,1,3,3)
{
    __shared__ float X[NCH * CHSZ];    // 43,200 B of 320 KB WGP LDS

    const int tid = threadIdx.x;
    const int wgx = blockIdx.x * TILE;
    const int wgy = blockIdx.y * TILE;

    // Uniform scalars -> SGPRs
    float Pm[9];
#pragma unroll
    for (int i = 0; i < 9; ++i) Pm[i] = P[i];
    const float cb0 = conv_b[0], cb1 = conv_b[1], cb2 = conv_b[2];

    // ---------------- Phase 1: 18x18 halo of (Z, w_pre) into LDS --------------
#pragma unroll
    for (int it = 0; it < 2; ++it) {
        int hp = tid + it * 256;
        if (hp < HALO * HALO) {
            const int hy = hp / HALO, hx = hp % HALO;
            const int gy = wgy + hy - 1, gx = wgx + hx - 1;
            const int lb = hy * PITCH + hx;
            if (gy >= 0 && gy < IMG && gx >= 0 && gx < IMG) {
                const int pix = gy * IMG + gx;
                const float o0 = obs[pix * 3 + 0], o1 = obs[pix * 3 + 1], o2 = obs[pix * 3 + 2];
                const float u0 = u_k[pix * 3 + 0], u1 = u_k[pix * 3 + 1], u2 = u_k[pix * 3 + 2];
                const float b0 = Pm[0] * o0 + Pm[3] * o1 + Pm[6] * o2;
                const float b1 = Pm[1] * o0 + Pm[4] * o1 + Pm[7] * o2;
                const float b2 = Pm[2] * o0 + Pm[5] * o1 + Pm[8] * o2;
                const float Bu0 = b0 * u0, Bu1 = b1 * u1, Bu2 = b2 * u2;
                const float bu  = Bu0 + Bu1 + Bu2;
                const float inv = 1.0f / bu;
                const float Z0 = Bu0 * inv, Z1 = Bu1 * inv, Z2 = Bu2 * inv;
                X[0 * CHSZ + lb] = Z0;
                X[1 * CHSZ + lb] = Z1;
                X[2 * CHSZ + lb] = Z2;
                const float ov[3] = {o0, o1, o2};
                const float uv[3] = {u0, u1, u2};
                const float bv[3] = {b0, b1, b2};
                const float Zv[3] = {Z0, Z1, Z2};
                const float* wp = w_k + (long)pix * 27;
#pragma unroll
                for (int ij = 0; ij < 9; ++ij) {
                    const int i = ij / 3, j = ij % 3;
                    // du_s = obs_i*u_s*delta(s==j) + b_s*w[s,i,j]
                    float du0 = bv[0] * wp[0 * 9 + ij] + ((j == 0) ? ov[i] * uv[0] : 0.0f);
                    float du1 = bv[1] * wp[1 * 9 + ij] + ((j == 1) ? ov[i] * uv[1] : 0.0f);
                    float du2 = bv[2] * wp[2 * 9 + ij] + ((j == 2) ? ov[i] * uv[2] : 0.0f);
                    float sdu = du0 + du1 + du2;
                    // (du*bu - sdu*Bu_s)/bu^2 == (du - sdu*Z_s)/bu
                    X[(3 + ij * 3 + 0) * CHSZ + lb] = (du0 - sdu * Zv[0]) * inv;
                    X[(3 + ij * 3 + 1) * CHSZ + lb] = (du1 - sdu * Zv[1]) * inv;
                    X[(3 + ij * 3 + 2) * CHSZ + lb] = (du2 - sdu * Zv[2]) * inv;
                }
            } else {
                // outside image: SAME padding -> zero contribution
#pragma unroll
                for (int c = 0; c < NCH; ++c) X[c * CHSZ + lb] = 0.0f;
            }
        }
    }
    __syncthreads();

    // ---------------- Phase 2: conv via V_WMMA_F32_16X16X4_F32 ----------------
    const int lane = tid & 31;
    const int wv   = tid >> 5;          // 8 waves, 2 pixel-rows each
    const int half = (lane < 16) ? 0 : 2; // A/B K-slot offset for upper half-wave
    const int n    = lane & 15;         // pixel column (N) / A row (M)

    // A = Wmat^T padded to 16x27 (rows>=3 zero): A[m=c_out, k=tap*3+cin]
    v2f Amat[7];
    int off0[7], off1[7];               // per-lane LDS element offsets for B gathers
#pragma unroll
    for (int jc = 0; jc < 7; ++jc) {
        const int k0 = jc * 4 + half;
        const int k1 = k0 + 1;
        float a0 = 0.0f, a1 = 0.0f;
        if (n < 3 && k0 < 27) a0 = conv_w[n * 27 + (k0 % 3) * 9 + (k0 / 3)];
        if (n < 3 && k1 < 27) a1 = conv_w[n * 27 + (k1 % 3) * 9 + (k1 / 3)];
        Amat[jc][0] = a0; Amat[jc][1] = a1;
        const int c0 = (k0 < 27) ? k0 : 26;   // clamp (A is zero there; avoid OOB/NaN)
        const int c1 = (k1 < 27) ? k1 : 26;
        const int t0 = c0 / 3, t1 = c1 / 3;
        off0[jc] = (c0 % 3) * CHSZ + (t0 / 3) * PITCH + (t0 % 3) + n;
        off1[jc] = (c1 % 3) * CHSZ + (t1 / 3) * PITCH + (t1 % 3) + n;
    }

    // one group's 3x3x3->3 conv for 16 pixels: D(16x16) = A(16x4k) @ B(4kx16)
    auto conv_group = [&](int gbase) -> v8f {
        v8f acc = {0.f, 0.f, 0.f, 0.f, 0.f, 0.f, 0.f, 0.f};
#pragma unroll
        for (int jc = 0; jc < 7; ++jc) {
            v2f Bv;
            Bv[0] = X[gbase + off0[jc]];
            Bv[1] = X[gbase + off1[jc]];
            acc = __builtin_amdgcn_wmma_f32_16x16x4_f32(
                false, Amat[jc], false, Bv, (short)0, acc, false, false);
        }
        return acc;
    };

#pragma unroll
    for (int r = 0; r < 2; ++r) {
        const int ty     = wv * 2 + r;
        const int gy     = wgy + ty;
        const int rowoff = ty * PITCH;

        // group 0: u_kp1 = softmax(conv(Z) + bias)
        v8f y = conv_group(rowoff);
        float y0 = y[0] + cb0, y1 = y[1] + cb1, y2 = y[2] + cb2;
        float m  = fmaxf(y0, fmaxf(y1, y2));
        float e0 = __expf(y0 - m), e1 = __expf(y1 - m), e2 = __expf(y2 - m);
        float is = 1.0f / (e0 + e1 + e2);
        float p0 = e0 * is, p1 = e1 * is, p2 = e2 * is;

        const int pix = gy * IMG + (wgx + n);
        if (lane < 16) {                       // lanes 0..15 hold pixels 0..15
            u_out[pix * 3 + 0] = p0;
            u_out[pix * 3 + 1] = p1;
            u_out[pix * 3 + 2] = p2;
        }

        // groups 1..9: softmax JVP  w_kp1 = p ⊙ (dy - <p,dy>)
        for (int g = 1; g < 10; ++g) {
            v8f dv = conv_group(g * (3 * CHSZ) + rowoff);
            float d0 = dv[0], d1 = dv[1], d2 = dv[2];
            float dot = p0 * d0 + p1 * d1 + p2 * d2;
            float w0 = p0 * (d0 - dot);
            float w1 = p1 * (d1 - dot);
            float w2 = p2 * (d2 - dot);
            if (lane < 16) {
                float* wo = w_out + (long)pix * 27 + (g - 1);
                wo[0]  = w0;   // s=0
                wo[9]  = w1;   // s=1
                wo[18] = w2;   // s=2
            }
        }
    }
}

extern "C" void kernel_launch(void* const* d_in, const int* in_sizes, int n_in,
                              void* d_out, int out_size, void* d_ws, size_t ws_size,
                              hipStream_t stream) {
    (void)in_sizes; (void)n_in; (void)out_size; (void)d_ws; (void)ws_size;
    const float* obs    = (const float*)d_in[0];
    const float* P      = (const float*)d_in[1];
    const float* u_k    = (const float*)d_in[2];
    const float* w_k    = (const float*)d_in[3];
    const float* conv_w = (const float*)d_in[4];
    const float* conv_b = (const float*)d_in[5];
    float* u_out = (float*)d_out;                    // (512,512,3,1)
    float* w_out = u_out + (size_t)IMG * IMG * 3;    // (512,512,3,1,3,3)

    dim3 grid(IMG / TILE, IMG / TILE);
    dim3 block(256);
    hmm_fused_kernel<<<grid, block, 0, stream>>>(obs, P, u_k, w_k, conv_w, conv_b,
                                                 u_out, w_out);
}